// MeshDecoder_17566416240839
// MI455X (gfx1250) — compile-verified
//
#include <hip/hip_runtime.h>
#include <cstdint>

#ifndef __has_builtin
#define __has_builtin(x) 0
#endif

// ---- problem geometry (GRID = 128) --------------------------------------
#define G      128
#define GM1    127
#define PLANE  (G * G)            // 16384   (+x neighbor stride in field)
#define YSTR   (GM1 * G)          // 16256   (x-stride of y/z edge arrays)
#define NX_E   (GM1 * G * G)      // 2080768 edges per axis
#define NEDGES (3 * NX_E)         // 6242304 total edge vertices
#define VERTF  (NEDGES * 3)       // 18726912 floats of verts output
#define NC     (GM1 * GM1 * GM1)  // 2048383 cells
#define Y_OFF  (NX_E)
#define Z_OFF  (2 * NX_E)

// int4 vector type matching the builtin's expected pointee
// ('__attribute__((__vector_size__(4 * sizeof(int)))) int' per the
// compiler diagnostic).
typedef int v4i_ __attribute__((vector_size(16)));

// ---- CDNA5 async LDS->global store path ---------------------------------
// One wave instruction moves 16B per lane straight from LDS to global
// memory, tracked by ASYNCcnt (CDNA5 ISA).
__device__ __forceinline__ void async_store_b128(float* gaddr, float* lds) {
#if __has_builtin(__builtin_amdgcn_global_store_async_from_lds_b128)
  __builtin_amdgcn_global_store_async_from_lds_b128(
      (__attribute__((address_space(1))) v4i_*)gaddr,
      (__attribute__((address_space(3))) v4i_*)lds,
      /*imm offset*/ 0, /*cpol*/ 0);
#else
  unsigned long long ga = (unsigned long long)gaddr;
  unsigned int la = (unsigned int)(unsigned long long)lds;  // low 32b = LDS offset
  asm volatile("global_store_async_from_lds_b128 %0, %1, off"
               :: "v"(ga), "v"(la) : "memory");
#endif
}

__device__ __forceinline__ void wait_async0() {
#if __has_builtin(__builtin_amdgcn_s_wait_asynccnt)
  __builtin_amdgcn_s_wait_asynccnt(0);
#else
  asm volatile("s_wait_asynccnt 0" ::: "memory");
#endif
}

// =========================================================================
// Kernel 1: edge vertices.  One thread per edge; NEDGES is an exact
// multiple of 256 and each axis region is an exact multiple of 256, so
// every block is full and its 768-float output range is 16B aligned.
// =========================================================================
__global__ void __launch_bounds__(256) verts_kernel(
    const float* __restrict__ field,
    const float* __restrict__ minp,
    const float* __restrict__ sizep,
    float* __restrict__ out) {
  __shared__ float stage[256 * 3];

  const int t = threadIdx.x;
  int e = blockIdx.x * 256 + t;  // < NEDGES by construction

  const float mn0 = minp[0], mn1 = minp[1], mn2 = minp[2];
  const float sz0 = sizep[0], sz1 = sizep[1], sz2 = sizep[2];
  const float inv = 1.0f / 127.0f;

  int axis, i, j, k, fi, off2;
  if (e < NX_E) {                       // x-edges: shape (127,128,128)
    axis = 0;
    i = e / PLANE; int r = e - i * PLANE; j = r >> 7; k = r & 127;
    fi = i * PLANE + j * G + k; off2 = PLANE;
  } else if (e < 2 * NX_E) {            // y-edges: shape (128,127,128)
    e -= NX_E; axis = 1;
    i = e / YSTR; int r = e - i * YSTR; j = r >> 7; k = r & 127;
    fi = i * PLANE + j * G + k; off2 = G;
  } else {                              // z-edges: shape (128,128,127)
    e -= 2 * NX_E; axis = 2;
    i = e / YSTR; int r = e - i * YSTR; j = r / GM1; k = r - j * GM1;
    fi = i * PLANE + j * G + k; off2 = 1;
  }

  const float s1 = field[fi];
  const float s2 = field[fi + off2];
  float d = s1 - s2;
  d = (d == 0.0f) ? 1.0f : d;
  const float tt = s1 / d;

  const float px = fmaf((float)i * inv, sz0, mn0);
  const float py = fmaf((float)j * inv, sz1, mn1);
  const float pz = fmaf((float)k * inv, sz2, mn2);
  const float dx = (axis == 0) ? sz0 * inv : 0.0f;
  const float dy = (axis == 1) ? sz1 * inv : 0.0f;
  const float dz = (axis == 2) ? sz2 * inv : 0.0f;

  stage[t * 3 + 0] = fmaf(tt, dx, px);
  stage[t * 3 + 1] = fmaf(tt, dy, py);
  stage[t * 3 + 2] = fmaf(tt, dz, pz);
  __syncthreads();

  // 768 floats = 192 aligned b128 chunks, streamed LDS -> HBM.
  if (t < 192) {
    async_store_b128(out + (size_t)blockIdx.x * 768 + t * 4, &stage[t * 4]);
  }
  wait_async0();
}

// =========================================================================
// Kernel 2: faces + face_mask.  One thread per cell.  Edge-index table in
// LDS (stride 13: gcd(13,64)=1 -> low bank conflict) so the dynamic
// cases-gather is a ds_load, not scratch.  Face floats staged in LDS in
// exact global order, flushed with async b128 stores (full blocks).
// =========================================================================
__global__ void __launch_bounds__(256) faces_kernel(
    const float* __restrict__ field,
    const int* __restrict__ cases,          // 256 x 5 x 3
    const unsigned char* __restrict__ masks,// 256 x 5 (bool)
    float* __restrict__ out) {
  __shared__ int   ledge[256 * 13];
  __shared__ float fstage[256 * 15];

  const int t  = threadIdx.x;
  const int c0 = blockIdx.x * 256;
  const int c  = c0 + t;
  const bool full  = (NC - c0) >= 256;   // uniform per block
  const bool valid = c < NC;

  float ftri[15];
  float fm[5];

  if (valid) {
    const int x = c / (GM1 * GM1);
    int r = c - x * (GM1 * GM1);
    const int y = r / GM1;
    const int z = r - y * GM1;

    // ---- 8-corner sign code (bit order from the reference's shifts) ----
    const int b = x * PLANE + y * G + z;
    int bm = 0;
    bm |= (field[b]                 > 0.0f) ? 1   : 0;  // (x  ,y  ,z  )
    bm |= (field[b + PLANE]         > 0.0f) ? 2   : 0;  // (x+1,y  ,z  )
    bm |= (field[b + G]             > 0.0f) ? 4   : 0;  // (x  ,y+1,z  )
    bm |= (field[b + PLANE + G]     > 0.0f) ? 8   : 0;  // (x+1,y+1,z  )
    bm |= (field[b + 1]             > 0.0f) ? 16  : 0;  // (x  ,y  ,z+1)
    bm |= (field[b + PLANE + 1]     > 0.0f) ? 32  : 0;  // (x+1,y  ,z+1)
    bm |= (field[b + G + 1]         > 0.0f) ? 64  : 0;  // (x  ,y+1,z+1)
    bm |= (field[b + PLANE + G + 1] > 0.0f) ? 128 : 0;  // (x+1,y+1,z+1)

    // ---- 12 global edge indices into LDS -------------------------------
    const int e0 = x * PLANE + y * G + z;
    const int e4 = Y_OFF + x * YSTR + y * G + z;
    const int e8 = Z_OFF + x * YSTR + y * GM1 + z;
    int* le = &ledge[t * 13];
    le[0]  = e0;            le[1]  = e0 + G;
    le[2]  = e0 + 1;        le[3]  = e0 + G + 1;
    le[4]  = e4;            le[5]  = e4 + YSTR;
    le[6]  = e4 + 1;        le[7]  = e4 + YSTR + 1;
    le[8]  = e8;            le[9]  = e8 + YSTR;
    le[10] = e8 + GM1;      le[11] = e8 + YSTR + GM1;

    // ---- case table gather ---------------------------------------------
    const int* tri = cases + bm * 15;
    const unsigned char* mk = masks + (size_t)bm * 5;
#pragma unroll
    for (int j = 0; j < 15; ++j) {
      ftri[j] = (float)le[tri[j]];   // ds_load, dynamic index 0..11
    }
#pragma unroll
    for (int j = 0; j < 5; ++j) fm[j] = mk[j] ? 1.0f : 0.0f;
  }

  // ---- face_mask: region base is only 4B-aligned -> plain b32 stores ----
  if (valid) {
    float* m = out + (size_t)VERTF + (size_t)NC * 15 + (size_t)c * 5;
#pragma unroll
    for (int j = 0; j < 5; ++j) m[j] = fm[j];
  }

  // ---- faces ------------------------------------------------------------
  if (full) {
#pragma unroll
    for (int j = 0; j < 15; ++j) fstage[t * 15 + j] = ftri[j];
    __syncthreads();
    // 256*15 = 3840 floats = 960 aligned b128 chunks (base % 4 floats == 0)
    float* fbase = out + (size_t)VERTF + (size_t)c0 * 15;
    for (int i = t; i < 960; i += 256) {
      async_store_b128(fbase + i * 4, &fstage[i * 4]);
    }
  } else if (valid) {  // single tail block (127 cells): direct stores
    float* f = out + (size_t)VERTF + (size_t)c * 15;
#pragma unroll
    for (int j = 0; j < 15; ++j) f[j] = ftri[j];
  }
  wait_async0();
}

// =========================================================================
extern "C" void kernel_launch(void* const* d_in, const int* in_sizes, int n_in,
                              void* d_out, int out_size, void* d_ws, size_t ws_size,
                              hipStream_t stream) {
  (void)in_sizes; (void)n_in; (void)out_size; (void)d_ws; (void)ws_size;
  const float*         field = (const float*)d_in[0];
  const float*         minp  = (const float*)d_in[1];
  const float*         sizep = (const float*)d_in[2];
  const int*           cases = (const int*)d_in[3];
  const unsigned char* masks = (const unsigned char*)d_in[4];
  float* out = (float*)d_out;

  verts_kernel<<<NEDGES / 256, 256, 0, stream>>>(field, minp, sizep, out);
  faces_kernel<<<(NC + 255) / 256, 256, 0, stream>>>(field, cases, masks, out);
}